// ExtractModel_67525475827889
// MI455X (gfx1250) — compile-verified
//
#include <hip/hip_runtime.h>
#include <hip/hip_bf16.h>
#include <cstdint>
#include <cfloat>
#include <math.h>

// ---------------------------------------------------------------------------
// Problem constants (match reference)
// ---------------------------------------------------------------------------
namespace {

constexpr int   kVocab   = 30000;
constexpr int   kBatch   = 8;
constexpr int   kLenS    = 50;
constexpr int   kLenE    = 10;
constexpr int   kSpans   = kLenS * kLenE;          // 500
constexpr int   kRows    = kBatch * kSpans;        // 4000 candidate spans
constexpr int   kTPB     = 256;                    // 8 waves (wave32)
constexpr int   kChunkF  = 4096;                   // floats per LDS buffer (16 KB)
constexpr int   kChunkF4 = kChunkF / 4;            // 1024 float4 per chunk
constexpr int   kNChunks = (kVocab + kChunkF - 1) / kChunkF;  // 8 (last = 1328 floats)
constexpr float kInvT    = 10.0f;                  // 1 / TEMPERATURE
constexpr float kThr     = 0.05f;                  // init_threshold

typedef __attribute__((ext_vector_type(2))) float v2f;
typedef __attribute__((ext_vector_type(8))) float v8f;

#if defined(__has_builtin)
#if __has_builtin(__builtin_amdgcn_wmma_f32_16x16x4_f32)
#define HAVE_WMMA_F32X4 1
#endif
#endif
#ifndef HAVE_WMMA_F32X4
#define HAVE_WMMA_F32X4 0
#endif

// ---------------------------------------------------------------------------
// CDNA5 async global->LDS DMA helpers (tracked by ASYNCcnt)
//   global_load_async_to_lds_b128  vdst(lds byte offset), vaddr(64b), off
// ---------------------------------------------------------------------------
__device__ __forceinline__ void async_cp_b128(uint32_t lds_off, const void* gaddr) {
  asm volatile("global_load_async_to_lds_b128 %0, %1, off"
               :: "v"(lds_off), "v"(gaddr) : "memory");
}
__device__ __forceinline__ void wait_async_le4() {
  // <=4 outstanding: the 4 ops of the previous chunk have completed (in-order)
  asm volatile("s_wait_asynccnt 4" ::: "memory");
}
__device__ __forceinline__ void wait_async_0() {
  asm volatile("s_wait_asynccnt 0" ::: "memory");
}

// ---------------------------------------------------------------------------
// Phase 1: per-row streaming soft-min over vocab (HBM-bound, 480 MB total).
// One block per row; double-buffered async-to-LDS pipeline; WMMA f32 block sum.
// ---------------------------------------------------------------------------
__global__ __launch_bounds__(kTPB)
void softmin_rows(const float* __restrict__ ed,
                  const unsigned char* __restrict__ viable,
                  float* __restrict__ out,          // scores written at out[32 + row]
                  int* __restrict__ vocab_dense) {  // workspace, kRows ints
  __shared__ float buf[2][kChunkF];                 // 32 KB double buffer
  __shared__ float rm[kTPB], rz[kTPB], rs[kTPB];    // rz/rs reused as 16x16 tiles
  __shared__ int   ri[kTPB];

  const int row = blockIdx.x;
  const int t   = threadIdx.x;
  const float* rowp = ed + (size_t)row * kVocab;    // 120 KB per row, 16B aligned

  // online softmin state: m = running min, Z = sum exp((m-x)/T),
  // S = sum x*exp((m-x)/T), idx = argmin
  float m = FLT_MAX, Z = 0.0f, S = 0.0f;
  int   idx = 0;

  auto issue = [&](int c) {
    const int base_f4 = c * kChunkF4;
    float* b = buf[c & 1];
#pragma unroll
    for (int k = 0; k < 4; ++k) {
      const int f4  = t + k * kTPB;                 // 0..1023 within chunk
      int gf4 = base_f4 + f4;
      // clamp tail lanes to a valid address so every thread issues exactly
      // 4 ops/chunk -> per-wave ASYNCcnt is deterministic (4 per chunk)
      if (gf4 > kVocab / 4 - 1) gf4 = kVocab / 4 - 1;
      async_cp_b128((uint32_t)(uintptr_t)(b + 4 * f4), rowp + 4 * gf4);
    }
  };

  issue(0);
  for (int c = 0; c < kNChunks; ++c) {
    if (c + 1 < kNChunks) { issue(c + 1); wait_async_le4(); }
    else                  { wait_async_0(); }
    __syncthreads();                                // all waves' DMA landed

    const int rem = kVocab - c * kChunkF;
    const int nf4 = ((rem < kChunkF) ? rem : kChunkF) / 4;   // 1024 or 332
    const float4* b4 = (const float4*)buf[c & 1];
#pragma unroll
    for (int k = 0; k < 4; ++k) {
      const int f4 = t + k * kTPB;
      if (f4 < nf4) {
        const float4 v = b4[f4];
        const int gi = c * kChunkF + 4 * f4;
        const float xs[4] = {v.x, v.y, v.z, v.w};
#pragma unroll
        for (int e = 0; e < 4; ++e) {
          const float x = xs[e];
          if (x < m) {                              // rare: new running min
            const float sc = __expf((x - m) * kInvT);
            Z = Z * sc + 1.0f;
            S = S * sc + x;
            m = x; idx = gi + e;
          } else {                                  // common: 1 exp + 2 FMA
            const float w = __expf((m - x) * kInvT);
            Z += w; S += x * w;
          }
        }
      }
    }
    __syncthreads();                                // buffer free for re-issue
  }

  // ---- block reduction ----
  auto finalize = [&](float Zs, float Ss) {
    const float matched_ed = Ss / Zs;               // soft-min value
    // soft_threshold: (celu(1 - 2x/thr) + 1) / 2, celu(y)=y>0?y:expm1(min(y,0))
    const float y   = 1.0f - (2.0f / kThr) * matched_ed;
    const float cel = (y > 0.0f) ? y : expm1f(fminf(y, 0.0f));
    const float score = 0.5f * (cel + 1.0f);
    const bool  vi  = viable[row] != 0;
    out[32 + row]    = vi ? score : 0.0f;           // score_dense (flat == row)
    vocab_dense[row] = vi ? ri[0] : -1;
  };

  // 1) min + first-index argmin tree
  rm[t] = m; ri[t] = idx;
  __syncthreads();
  for (int off = kTPB / 2; off > 0; off >>= 1) {
    if (t < off) {
      const float ma = rm[t], mb = rm[t + off];
      const int  ia = ri[t], ib = ri[t + off];
      if (mb < ma || (mb == ma && ib < ia)) { rm[t] = mb; ri[t] = ib; }
    }
    __syncthreads();
  }
  const float mstar = rm[0];

  // 2) rescale partials to the global min -> remaining reduction is a plain sum
  //    (exactly equivalent to the pairwise online-softmax merge)
  const float f = __expf((mstar - m) * kInvT);
  rz[t] = Z * f;                                    // 16x16 tile, t = r*16 + c
  rs[t] = S * f;
  __syncthreads();

#if HAVE_WMMA_F32X4
  // 3) sum both 16x16 tiles on the matrix pipe: ones(16x4) x tile(4x16),
  //    accumulated over 4 K-chunks -> every lane of D vgpr0 holds a column sum.
  if (t < 32) {                                     // wave 0, EXEC all ones
    const int col  = t & 15;
    const int rsel = t >> 4;                        // B vgpr0: rows 4g,4g+1; vgpr1: 4g+2,4g+3
    const v2f A = {1.0f, 1.0f};                     // 16x4 all-ones A operand
    v8f accZ = {};
    v8f accS = {};
#pragma unroll
    for (int g = 0; g < 4; ++g) {
      v2f bz, bs;
      bz[0] = rz[(4 * g + rsel) * 16 + col];
      bz[1] = rz[(4 * g + 2 + rsel) * 16 + col];
      bs[0] = rs[(4 * g + rsel) * 16 + col];
      bs[1] = rs[(4 * g + 2 + rsel) * 16 + col];
      // v_wmma_f32_16x16x4_f32: D = A*B + C (full fp32, RNE)
      accZ = __builtin_amdgcn_wmma_f32_16x16x4_f32(false, A, false, bz,
                                                   (short)0, accZ, false, false);
      accS = __builtin_amdgcn_wmma_f32_16x16x4_f32(false, A, false, bs,
                                                   (short)0, accS, false, false);
    }
    float zc = accZ[0];                             // colsum[t & 15]
    float sc = accS[0];
#pragma unroll
    for (int off = 8; off >= 1; off >>= 1) {        // fold 16 column sums
      zc += __shfl_xor(zc, off, 32);
      sc += __shfl_xor(sc, off, 32);
    }
    if (t == 0) finalize(zc, sc);
  }
#else
  for (int off = kTPB / 2; off > 0; off >>= 1) {
    if (t < off) { rz[t] += rz[t + off]; rs[t] += rs[t + off]; }
    __syncthreads();
  }
  if (t == 0) finalize(rz[0], rs[0]);
#endif
}

// ---------------------------------------------------------------------------
// Phase 2: per-batch soft-max over 500 spans (tiny). One block per batch.
// ---------------------------------------------------------------------------
__global__ __launch_bounds__(512)
void softmax_spans(const float* __restrict__ scores,      // = out + 32
                   const int* __restrict__ vocab_dense,
                   float* __restrict__ out) {
  __shared__ float mv[512]; __shared__ int mi[512];
  __shared__ float zz[512]; __shared__ float ss[512];
  const int b = blockIdx.x, t = threadIdx.x;
  const bool valid = (t < kSpans);
  const float x = valid ? scores[b * kSpans + t] : -FLT_MAX;

  // max + first-index argmax
  mv[t] = x; mi[t] = valid ? t : 0x7FFFFFFF;
  __syncthreads();
  for (int off = 256; off > 0; off >>= 1) {
    if (t < off) {
      const float a = mv[t], c = mv[t + off];
      const int  ia = mi[t], ic = mi[t + off];
      if (c > a || (c == a && ic < ia)) { mv[t] = c; mi[t] = ic; }
    }
    __syncthreads();
  }
  const float mx  = mv[0];
  const int  bidx = mi[0];
  __syncthreads();

  // stable softmax(x/T) weighted sum over exactly the 500 valid entries
  const float w = valid ? __expf((x - mx) * kInvT) : 0.0f;
  zz[t] = w;
  ss[t] = valid ? x * w : 0.0f;
  __syncthreads();
  for (int off = 256; off > 0; off >>= 1) {
    if (t < off) { zz[t] += zz[t + off]; ss[t] += ss[t + off]; }
    __syncthreads();
  }

  if (t == 0) {
    const float val  = ss[0] / zz[0];
    const int start  = bidx / kLenE;
    const int rem    = bidx % kLenE;
    out[b]      = val;                              // best_matched_score
    out[8 + b]  = (float)start;                     // start
    out[16 + b] = (float)(start + rem);             // end
    out[24 + b] = (float)vocab_dense[b * kSpans + bidx];  // best_matched_vocab
  }
}

} // namespace

// ---------------------------------------------------------------------------
// Entry point
//   d_in[0] ed_dist (N*VOCAB f32), d_in[4] viable (N bool bytes)
//   d_out: [0:8) score, [8:16) start, [16:24) end, [24:32) vocab,
//          [32:4032) score_dense
// ---------------------------------------------------------------------------
extern "C" void kernel_launch(void* const* d_in, const int* in_sizes, int n_in,
                              void* d_out, int out_size, void* d_ws, size_t ws_size,
                              hipStream_t stream) {
  (void)in_sizes; (void)n_in; (void)out_size; (void)ws_size;
  const float*         ed     = (const float*)d_in[0];
  const unsigned char* viable = (const unsigned char*)d_in[4];
  float* out      = (float*)d_out;
  int*   vocab_ws = (int*)d_ws;                     // kRows ints of scratch

  softmin_rows<<<dim3(kRows), dim3(kTPB), 0, stream>>>(ed, viable, out, vocab_ws);
  softmax_spans<<<dim3(kBatch), dim3(512), 0, stream>>>(out + 32, vocab_ws, out);
}